// DeepGIN_14302241096098
// MI455X (gfx1250) — compile-verified
//
#include <hip/hip_runtime.h>

#define N_NODES 100000
#define HIDDEN  128
#define DEPTH   4
#define EDGES   625000
#define BN_EPS  1e-5f

typedef float v2f __attribute__((ext_vector_type(2)));
typedef float v8f __attribute__((ext_vector_type(8)));

// ---------------------------------------------------------------------------
// tmp[i] = (1 + eps[layer]) * h[i]           (vectorized elementwise)
// ---------------------------------------------------------------------------
__global__ __launch_bounds__(256) void gin_scale_kernel(
    const float* __restrict__ h, const float* __restrict__ eps, int layer,
    float* __restrict__ out) {
  const float s = 1.0f + eps[layer];
  const int n4 = N_NODES * HIDDEN / 4;  // 3,200,000 float4
  int i = blockIdx.x * blockDim.x + threadIdx.x;
  if (i < n4) {
    float4 v = ((const float4*)h)[i];
    ((float4*)out)[i] = make_float4(s * v.x, s * v.y, s * v.z, s * v.w);
  }
}

// ---------------------------------------------------------------------------
// agg[dst] += h[src] for every edge; one thread per (edge, float4 chunk).
// Coalesced gather + coalesced hardware float atomics (global_atomic_add_f32).
// ---------------------------------------------------------------------------
__global__ __launch_bounds__(256) void gin_scatter_kernel(
    const float* __restrict__ h, const long long* __restrict__ ei,
    float* __restrict__ agg) {
  unsigned int gid = blockIdx.x * blockDim.x + threadIdx.x;
  unsigned int e = gid >> 5;  // edge id
  unsigned int c = gid & 31;  // float4 chunk within the 128-wide row
  if (e >= EDGES) return;
  int src = (int)ei[e];
  int dst = (int)ei[EDGES + e];
  float4 v = ((const float4*)(h + (long long)src * HIDDEN))[c];
  float* p = agg + (long long)dst * HIDDEN + c * 4;
  unsafeAtomicAdd(p + 0, v.x);
  unsafeAtomicAdd(p + 1, v.y);
  unsafeAtomicAdd(p + 2, v.z);
  unsafeAtomicAdd(p + 3, v.w);
}

// ---------------------------------------------------------------------------
// Fused GEMM + epilogue:  out = act(A @ W^T + bias)
//   STAGE 0 : act = ReLU                  (MLP hidden layer)
//   STAGE 1 : act = ELU(BatchNorm(.))     (MLP out + BN + ELU)
//
// Block = 256 threads = 8 waves; block tile = 128 rows x 128 cols.
// W^T staged in LDS in k-PAIRED layout:
//     lds_wp[(k>>1)*128 + n] = { W[n][k], W[n][k+1] }          (64 KB)
// so each WMMA B fragment is one aligned ds_load_b64 that lands directly in
// an even VGPR pair (V_WMMA_F32_16X16X4_F32: lanes 0-15 -> K {0,1}, lanes
// 16-31 -> K {2,3}; kb is 4-aligned, koff even -> pair index always integral).
// Staging uses GLOBAL_LOAD_ASYNC_TO_LDS_B32 (ASYNCcnt) when the builtin is
// available: the per-lane LDS address performs the transpose during the DMA.
// ---------------------------------------------------------------------------
template <int STAGE>
__global__ __launch_bounds__(256) void gin_gemm_kernel(
    const float* __restrict__ A,      // [N_NODES, 128]
    const float* __restrict__ W,      // [128, 128], W[out][in]
    const float* __restrict__ bias,   // [128]
    const float* __restrict__ gamma,  // [128] (STAGE 1 only)
    const float* __restrict__ beta,   // [128]
    const float* __restrict__ mean,   // [128]
    const float* __restrict__ var,    // [128]
    float* __restrict__ out) {
  __shared__ v2f lds_wp[(HIDDEN / 2) * HIDDEN];  // 64 KB paired W^T

  const int tid = threadIdx.x;
  // Stage W transposed+paired. Global reads coalesced along k (i = n*128+k).
#if __has_builtin(__builtin_amdgcn_global_load_async_to_lds_b32)
#pragma unroll 8
  for (int i = tid; i < HIDDEN * HIDDEN; i += 256) {
    int k = i & 127;
    int n = i >> 7;
    const float* gsrc = W + i;  // W[n][k]
    float* ldst = ((float*)lds_wp) + (((k >> 1) * HIDDEN + n) << 1) + (k & 1);
    __builtin_amdgcn_global_load_async_to_lds_b32(
        (__attribute__((address_space(1))) int*)gsrc,
        (__attribute__((address_space(3))) int*)ldst, 0, 0);
  }
#if __has_builtin(__builtin_amdgcn_s_wait_asynccnt)
  __builtin_amdgcn_s_wait_asynccnt(0);
#else
  asm volatile("s_wait_asynccnt 0x0" ::: "memory");
#endif
#else
#pragma unroll 8
  for (int i = tid; i < HIDDEN * HIDDEN; i += 256) {
    int k = i & 127;
    int n = i >> 7;
    ((float*)lds_wp)[(((k >> 1) * HIDDEN + n) << 1) + (k & 1)] = W[i];
  }
#endif
  __syncthreads();

  const int wave = tid >> 5;
  const int lane = tid & 31;
  const int m    = lane & 15;         // A row within 16-tile / B col in tile
  const int koff = (lane >> 4) << 1;  // 0 (lanes 0-15) or 2 (lanes 16-31)
  const int rowTile = blockIdx.x * 128 + wave * 16;

  // A-fragment source row (clamped so EXEC stays all-ones through WMMA).
  int arow = rowTile + m;
  if (arow >= N_NODES) arow = N_NODES - 1;
  const float* aptr = A + (long long)arow * HIDDEN + koff;

  // Bias folded into accumulator init: C[m][n] starts at bias[n].
  v8f acc[8];
#pragma unroll
  for (int nt = 0; nt < 8; ++nt) {
    float bv = bias[nt * 16 + m];
    acc[nt] = (v8f){bv, bv, bv, bv, bv, bv, bv, bv};
  }

  // K loop: one v2f A-fragment load feeds 8 column-tile WMMAs; each B
  // fragment is a single aligned ds_load_b64 from the paired layout.
#pragma unroll 4
  for (int kb = 0; kb < HIDDEN; kb += 4) {
    v2f a = *(const v2f*)(aptr + kb);  // A[m][kb+koff], A[m][kb+koff+1]
    const v2f* wrow = lds_wp + ((kb + koff) >> 1) * HIDDEN + m;
#pragma unroll
    for (int nt = 0; nt < 8; ++nt) {
      v2f b = wrow[nt * 16];  // { W[n][kb+koff], W[n][kb+koff+1] }, n=nt*16+m
      acc[nt] = __builtin_amdgcn_wmma_f32_16x16x4_f32(
          false, a, false, b, (short)0, acc[nt], false, false);
    }
  }

  // Epilogue + store. C layout: VGPR j, lane L -> M = j + 8*(L>=16), N = L&15.
  const int mhi = (lane >> 4) << 3;  // 0 or 8
#pragma unroll
  for (int nt = 0; nt < 8; ++nt) {
    int col = nt * 16 + m;
    float scale = 1.0f, shift = 0.0f;
    if (STAGE == 1) {
      scale = gamma[col] * rsqrtf(var[col] + BN_EPS);
      shift = beta[col] - mean[col] * scale;
    }
#pragma unroll
    for (int j = 0; j < 8; ++j) {
      int row = rowTile + mhi + j;
      if (row < N_NODES) {
        float y = acc[nt][j];
        if (STAGE == 0) {
          y = fmaxf(y, 0.0f);                       // ReLU
        } else {
          y = y * scale + shift;                    // BatchNorm (eval)
          y = (y > 0.0f) ? y : (__expf(y) - 1.0f);  // ELU
        }
        out[(long long)row * HIDDEN + col] = y;
      }
    }
  }
}

// ---------------------------------------------------------------------------
extern "C" void kernel_launch(void* const* d_in, const int* in_sizes, int n_in,
                              void* d_out, int out_size, void* d_ws,
                              size_t ws_size, hipStream_t stream) {
  const float*     x     = (const float*)d_in[0];
  const long long* ei    = (const long long*)d_in[1];
  const float*     eps   = (const float*)d_in[2];
  const float*     W1    = (const float*)d_in[3];
  const float*     b1    = (const float*)d_in[4];
  const float*     W2    = (const float*)d_in[5];
  const float*     b2    = (const float*)d_in[6];
  const float*     gamma = (const float*)d_in[7];
  const float*     beta  = (const float*)d_in[8];
  const float*     mean  = (const float*)d_in[9];
  const float*     var   = (const float*)d_in[10];
  float*           out   = (float*)d_out;

  const size_t NH = (size_t)N_NODES * HIDDEN;
  float* tmp  = (float*)d_ws;  // (1+eps)h + agg
  float* mid  = tmp + NH;      // relu(h W1^T + b1)
  float* hbuf = mid + NH;      // layer output ping buffer

  const dim3 blk(256);
  const int gridScale   = (N_NODES * HIDDEN / 4 + 255) / 256;
  const int gridScatter = (int)(((long long)EDGES * 32 + 255) / 256);
  const int gridGemm    = (N_NODES + 127) / 128;

  const float* h = x;
  for (int l = 0; l < DEPTH; ++l) {
    const float* W1l = W1 + (size_t)l * HIDDEN * HIDDEN;
    const float* W2l = W2 + (size_t)l * HIDDEN * HIDDEN;

    gin_scale_kernel<<<gridScale, blk, 0, stream>>>(h, eps, l, tmp);
    gin_scatter_kernel<<<gridScatter, blk, 0, stream>>>(h, ei, tmp);
    gin_gemm_kernel<0><<<gridGemm, blk, 0, stream>>>(
        tmp, W1l, b1 + l * HIDDEN, nullptr, nullptr, nullptr, nullptr, mid);
    float* hn = (l == DEPTH - 1) ? out : hbuf;
    gin_gemm_kernel<1><<<gridGemm, blk, 0, stream>>>(
        mid, W2l, b2 + l * HIDDEN, gamma + l * HIDDEN, beta + l * HIDDEN,
        mean + l * HIDDEN, var + l * HIDDEN, hn);
    h = hn;
  }
}